// Exact_58454504899045
// MI455X (gfx1250) — compile-verified
//
#include <hip/hip_runtime.h>
#include <math.h>

namespace {

constexpr int kB = 16, kE = 2000, kF = 8000, kQ = 20, kD = 100, kL = 3;
constexpr int kNumEntity = 500000;
constexpr int kNumWord = 200000;
constexpr float kLambda = 0.8f;
constexpr float kFactScale = 3.0f;
constexpr float kVeryNeg = -100000000000.0f;
constexpr float kVerySmall = 1e-10f;
constexpr float kDivInv = 0.1f; // 1/sqrt(D)
constexpr int kNP = 112;        // N padded to 7 tiles of 16

typedef float v2f __attribute__((ext_vector_type(2)));
typedef float v8f __attribute__((ext_vector_type(8)));

__device__ __forceinline__ float sigf(float x) { return 1.0f / (1.0f + __expf(-x)); }

// ---------------------------------------------------------------------------
// Zero-pad a (N x K) weight matrix to (112 x K) so GEMM B-loads need no masks.
// ---------------------------------------------------------------------------
__global__ void pad_weight_kernel(const float* __restrict__ W, float* __restrict__ Wp,
                                  int N, int K) {
  long i = (long)blockIdx.x * blockDim.x + threadIdx.x;
  if (i >= (long)kNP * K) return;
  int n = (int)(i / K);
  int k = (int)(i % K);
  Wp[i] = (n < N) ? W[(long)n * K + k] : 0.f;
}

// ---------------------------------------------------------------------------
// LSTM over the query: one block per batch element (B=16), 128 threads.
// ---------------------------------------------------------------------------
__global__ __launch_bounds__(128) void lstm_kernel(
    const int* __restrict__ qtext, const float* __restrict__ word_table,
    const float* __restrict__ wih, const float* __restrict__ whh,
    const float* __restrict__ bih, const float* __restrict__ bhh,
    float* __restrict__ q_hidden, float* __restrict__ q_last) {
  int b = blockIdx.x;
  __shared__ float xt[kD], h[kD], c[kD], g[4 * kD];
  int tid = threadIdx.x;
  for (int d = tid; d < kD; d += 128) { h[d] = 0.f; c[d] = 0.f; }
  __syncthreads();
  for (int t = 0; t < kQ; ++t) {
    int w = qtext[b * kQ + t];
    for (int d = tid; d < kD; d += 128) xt[d] = word_table[(long)w * kD + d];
    __syncthreads();
    for (int j = tid; j < 4 * kD; j += 128) {
      float acc = bih[j] + bhh[j];
      const float* wi = wih + (long)j * kD;
      const float* wh = whh + (long)j * kD;
      for (int k = 0; k < kD; ++k) acc += wi[k] * xt[k] + wh[k] * h[k];
      g[j] = acc;
    }
    __syncthreads();
    for (int d = tid; d < kD; d += 128) {
      float iv = sigf(g[d]);
      float fv = sigf(g[kD + d]);
      float gv = tanhf(g[2 * kD + d]);
      float ov = sigf(g[3 * kD + d]);
      float cn = fv * c[d] + iv * gv;
      float hn = ov * tanhf(cn);
      c[d] = cn;
      h[d] = hn;
      q_hidden[((long)b * kQ + t) * kD + d] = hn;
    }
    __syncthreads();
  }
  for (int d = tid; d < kD; d += 128) q_last[b * kD + d] = h[d];
}

// ---------------------------------------------------------------------------
// Fused attention: sim -> softmax(over q) -> att -> Wf.  One wave per (b,f).
// ---------------------------------------------------------------------------
__global__ __launch_bounds__(256) void attn_kernel(
    const float* __restrict__ q_hidden, const float* __restrict__ fact_emb,
    const int* __restrict__ qtext, float* __restrict__ Wf) {
  int wave = (blockIdx.x * blockDim.x + threadIdx.x) >> 5;
  int lane = threadIdx.x & 31;
  if (wave >= kB * kF) return;
  int b = wave / kF;
  const float* fr = fact_emb + (long)wave * kD;
  const float* qh = q_hidden + (long)b * kQ * kD;
  float fv[4], dm[4];
  for (int j = 0; j < 4; ++j) {
    int d = lane + 32 * j;
    dm[j] = (d < kD) ? 1.f : 0.f;
    fv[j] = (d < kD) ? fr[d] : 0.f;
  }
  float s[kQ];
  float smax = -3.0e38f;
  for (int q = 0; q < kQ; ++q) {
    float p = 0.f;
    for (int j = 0; j < 4; ++j) {
      int d = lane + 32 * j;
      int dc = d < kD ? d : 0;
      p += fv[j] * qh[q * kD + dc] * dm[j];
    }
    for (int off = 16; off; off >>= 1) p += __shfl_xor(p, off, 32);
    p *= kDivInv;
    float mask = (qtext[b * kQ + q] != kNumWord) ? 1.f : 0.f;
    p += (1.f - mask) * kVeryNeg;
    s[q] = p;
    smax = fmaxf(smax, p);
  }
  float ssum = 0.f;
  for (int q = 0; q < kQ; ++q) { s[q] = __expf(s[q] - smax); ssum += s[q]; }
  float inv = 1.f / ssum;
  float wf = 0.f;
  for (int j = 0; j < 4; ++j) {
    int d = lane + 32 * j;
    int dc = d < kD ? d : 0;
    float a = 0.f;
    for (int q = 0; q < kQ; ++q) a += s[q] * inv * qh[q * kD + dc];
    wf += a * fv[j] * dm[j];
  }
  for (int off = 16; off; off >>= 1) wf += __shfl_xor(wf, off, 32);
  if (lane == 0) Wf[wave] = wf * kDivInv;
}

// ---------------------------------------------------------------------------
// WMMA FP32 GEMM:  out[R,N] = A[R,K] @ Wp[112,K]^T + bias  (+ fused epilogues)
// Wp is the zero-padded weight (112 x K), so all B loads are unconditional and
// feed V_WMMA_F32_16X16X4_F32 directly.  One wave = one 16 x 112 strip; the A
// fragment is loaded once per K-step and reused by 7 independent WMMAs.
// ---------------------------------------------------------------------------
enum { EP_BIAS = 0, EP_RELU = 1, EP_E2F = 2, EP_F2E = 3 };

template <int MODE, bool AGATHER, int TN>
__global__ __launch_bounds__(256) void wmma_gemm(
    const float* __restrict__ A, const int* __restrict__ aidx,
    const float* __restrict__ Wp, const float* __restrict__ bias,
    float* __restrict__ out, int R, int N, int K,
    const int* __restrict__ kb_head, const float* __restrict__ gsrc,
    const float* __restrict__ wtilde, const float* __restrict__ prdiv,
    const float* __restrict__ addsrc, float addscale, int Fdim, int Edim) {
  int wave = (blockIdx.x * 256 + threadIdx.x) >> 5;
  int lane = threadIdx.x & 31;
  if (wave * 16 >= R) return;  // wave-uniform: EXEC stays all-ones
  int half = lane >> 4;        // 0: K={0,1}, 1: K={2,3}
  int lid = lane & 15;
  int mrow = wave * 16 + lid;
  const float* ap = A + (AGATHER ? (long)aidx[mrow] * K : (long)mrow * K);

  v8f z = {};
  v8f acc[TN];
#pragma unroll
  for (int j = 0; j < TN; ++j) acc[j] = z;

  for (int kk = 0; kk < K; kk += 4) {
    int kb = kk + 2 * half;
    v2f a;
    a.x = ap[kb];
    a.y = ap[kb + 1];
#pragma unroll
    for (int j = 0; j < TN; ++j) {
      int wo = (j * 16 + lid) * K + kb;  // padded: always in range
      v2f bv;
      bv.x = Wp[wo];
      bv.y = Wp[wo + 1];
      acc[j] = __builtin_amdgcn_wmma_f32_16x16x4_f32(
          false, a, false, bv, (short)0, acc[j], false, false);
    }
  }

#pragma unroll
  for (int j = 0; j < TN; ++j) {
    int ncol = j * 16 + lid;
    if (ncol >= N) continue;  // predicated epilogue; all WMMAs already done
    float bval = bias ? bias[ncol] : 0.f;
#pragma unroll
    for (int r = 0; r < 8; ++r) {
      int m = wave * 16 + r + 8 * half;  // C/D layout: VGPR r -> row r (+8 upper half)
      float val = acc[j][r] + bval;
      if (MODE == EP_RELU) {
        val = fmaxf(val, 0.f);
      } else if (MODE == EP_E2F) {
        int bidx = m / Fdim;
        int hh = kb_head[m];
        long ge = (long)bidx * Edim + hh;
        val += gsrc[ge * N + ncol];
        val = fmaxf(val, 0.f);
        val *= wtilde[m] * prdiv[ge];
      } else if (MODE == EP_F2E) {
        val += addscale * addsrc[(long)m * N + ncol];
        val = fmaxf(val, 0.f);
      }
      out[(long)m * N + ncol] = val;
    }
  }
}

// ---------------------------------------------------------------------------
// Small / elementwise / scatter kernels
// ---------------------------------------------------------------------------
__global__ void rowmax_kernel(const float* __restrict__ Wf, float* __restrict__ wmax) {
  int b = blockIdx.x;
  __shared__ float red[256];
  float m = -3.0e38f;
  for (int f = threadIdx.x; f < kF; f += 256) m = fmaxf(m, Wf[b * kF + f]);
  red[threadIdx.x] = m;
  __syncthreads();
  for (int s = 128; s; s >>= 1) {
    if (threadIdx.x < s) red[threadIdx.x] = fmaxf(red[threadIdx.x], red[threadIdx.x + s]);
    __syncthreads();
  }
  if (threadIdx.x == 0) wmax[b] = red[0];
}

__global__ void wtilde_kernel(const float* __restrict__ Wf, const float* __restrict__ wmax,
                              float* __restrict__ wt, int n) {
  int i = blockIdx.x * blockDim.x + threadIdx.x;
  if (i < n) wt[i] = __expf(Wf[i] - wmax[i / kF]);
}

__global__ void scatter_scalar_kernel(const float* __restrict__ src,
                                      const int* __restrict__ idx,
                                      float* __restrict__ dst, int n) {
  int i = blockIdx.x * blockDim.x + threadIdx.x;
  if (i >= n) return;
  int b = i / kF;
  atomicAdd(&dst[b * kE + idx[i]], src[i]);
}

__global__ void scatter_pr_kernel(const float* __restrict__ pg,
                                  const int* __restrict__ kb_head,
                                  const int* __restrict__ kb_tail,
                                  float* __restrict__ pr, int n) {
  int i = blockIdx.x * blockDim.x + threadIdx.x;
  if (i >= n) return;
  int b = i / kF;
  atomicAdd(&pr[b * kE + kb_tail[i]], pg[b * kE + kb_head[i]]);
}

__global__ void scatter_rows_kernel(const float* __restrict__ src,
                                    const int* __restrict__ kb_tail,
                                    float* __restrict__ dst, long n) {
  long i = (long)blockIdx.x * blockDim.x + threadIdx.x;
  if (i >= n) return;
  int d = (int)(i % kD);
  long m = i / kD;
  int b = (int)(m / kF);
  atomicAdd(&dst[((long)b * kE + kb_tail[m]) * kD + d], src[i]);
}

__global__ void clampmin_kernel(float* __restrict__ x, float lo, int n) {
  int i = blockIdx.x * blockDim.x + threadIdx.x;
  if (i < n) x[i] = fmaxf(x[i], lo);
}

__global__ void div_kernel(const float* __restrict__ a, const float* __restrict__ b,
                           float* __restrict__ o, int n) {
  int i = blockIdx.x * blockDim.x + threadIdx.x;
  if (i < n) o[i] = a[i] / b[i];
}

__global__ void copy_kernel(const float* __restrict__ s, float* __restrict__ d, long n) {
  long i = (long)blockIdx.x * blockDim.x + threadIdx.x;
  if (i < n) d[i] = s[i];
}

__global__ void pg_update_kernel(const float* __restrict__ pr, float* __restrict__ pg, int n) {
  int i = blockIdx.x * blockDim.x + threadIdx.x;
  if (i < n) pg[i] = kLambda * pr[i] + (1.f - kLambda) * pg[i];
}

__global__ void gather_entity_kernel(const int* __restrict__ le,
                                     const float* __restrict__ etab,
                                     float* __restrict__ out, long n) {
  long i = (long)blockIdx.x * blockDim.x + threadIdx.x;
  if (i >= n) return;
  int d = (int)(i % kD);
  long m = i / kD;
  out[i] = etab[(long)le[m] * kD + d];
}

__global__ void vecmat_kernel(const float* __restrict__ qn, const float* __restrict__ Wm,
                              const float* __restrict__ bb, float* __restrict__ outv) {
  int b = blockIdx.x;
  for (int d = threadIdx.x; d < kD; d += blockDim.x) {
    float acc = bb[d];
    for (int k = 0; k < kD; ++k) acc += qn[b * kD + k] * Wm[d * kD + k];
    outv[b * kD + d] = acc;
  }
}

__global__ void concat_kernel(const float* __restrict__ le, const float* __restrict__ q2ev,
                              const float* __restrict__ f2e, float* __restrict__ nxt, long n) {
  long i = (long)blockIdx.x * blockDim.x + threadIdx.x;
  if (i >= n) return;
  int col = (int)(i % (3 * kD));
  long m = i / (3 * kD);
  int b = (int)(m / kE);
  float v;
  if (col < kD) v = le[m * kD + col];
  else if (col < 2 * kD) v = q2ev[b * kD + (col - kD)];
  else v = f2e[m * kD + (col - 2 * kD)];
  nxt[i] = v;
}

__global__ void qn_reduce_kernel(const float* __restrict__ pg, const float* __restrict__ nq,
                                 float* __restrict__ qn) {
  int b = blockIdx.x / kD;
  int d = blockIdx.x % kD;
  __shared__ float red[256];
  float acc = 0.f;
  for (int e = threadIdx.x; e < kE; e += 256)
    acc += pg[b * kE + e] * nq[((long)b * kE + e) * kD + d];
  red[threadIdx.x] = acc;
  __syncthreads();
  for (int s = 128; s; s >>= 1) {
    if (threadIdx.x < s) red[threadIdx.x] += red[threadIdx.x + s];
    __syncthreads();
  }
  if (threadIdx.x == 0) qn[b * kD + d] = red[0];
}

__global__ void score_kernel(const float* __restrict__ emb, const float* __restrict__ sw,
                             const float* __restrict__ sb, const int* __restrict__ le,
                             float* __restrict__ out, int n) {
  int m = blockIdx.x * blockDim.x + threadIdx.x;
  if (m >= n) return;
  float acc = sb[0];
  const float* row = emb + (long)m * kD;
  for (int k = 0; k < kD; ++k) acc += row[k] * sw[k];
  if (le[m] == kNumEntity) acc += kVeryNeg;
  out[m] = acc;
}

inline int gemm_blocks(int R) {
  int waves = R / 16;          // one wave per 16-row strip (covers all N tiles)
  return (waves + 7) / 8;      // 8 waves per 256-thread block
}
inline int blks(long n, int t) { return (int)((n + t - 1) / t); }

}  // namespace

extern "C" void kernel_launch(void* const* d_in, const int* in_sizes, int n_in,
                              void* d_out, int out_size, void* d_ws, size_t ws_size,
                              hipStream_t stream) {
  (void)in_sizes; (void)n_in; (void)out_size; (void)ws_size;
  const int*   local_entity = (const int*)d_in[0];
  const int*   kb_fact_rel  = (const int*)d_in[1];
  const int*   kb_head      = (const int*)d_in[2];
  const int*   kb_tail      = (const int*)d_in[3];
  const int*   query_text   = (const int*)d_in[4];
  const float* q2e_adj      = (const float*)d_in[5];
  const float* entity_table = (const float*)d_in[6];
  const float* rel_table    = (const float*)d_in[7];
  const float* word_table   = (const float*)d_in[8];
  const float* rel_lin_w    = (const float*)d_in[9];
  const float* rel_lin_b    = (const float*)d_in[10];
  const float* lstm_wih     = (const float*)d_in[11];
  const float* lstm_whh     = (const float*)d_in[12];
  const float* lstm_bih     = (const float*)d_in[13];
  const float* lstm_bhh     = (const float*)d_in[14];
  const float* q2e_w        = (const float*)d_in[15];
  const float* q2e_b        = (const float*)d_in[16];
  const float* e2q_w        = (const float*)d_in[17];
  const float* e2q_b        = (const float*)d_in[18];
  const float* e2e_w        = (const float*)d_in[19];
  const float* e2e_b        = (const float*)d_in[20];
  const float* head_w       = (const float*)d_in[21];
  const float* head_b       = (const float*)d_in[22];
  const float* tail_w       = (const float*)d_in[23];
  const float* tail_b       = (const float*)d_in[24];
  const float* self_w       = (const float*)d_in[25];
  const float* self_b       = (const float*)d_in[26];
  const float* score_w      = (const float*)d_in[27];
  const float* score_b      = (const float*)d_in[28];
  float* out = (float*)d_out;

  const long BF = (long)kB * kF, BE = (long)kB * kE;
  float* ws = (float*)d_ws;
  size_t off = 0;
  float* fact_emb = ws + off; off += BF * kD;      // 12.8M
  float* e2f      = ws + off; off += BF * kD;      // 12.8M
  float* tailp    = ws + off; off += BF * kD;      // 12.8M
  float* nxt      = ws + off; off += BE * 3 * kD;  // 9.6M
  float* local    = ws + off; off += BE * kD;      // 3.2M
  float* headp    = ws + off; off += BE * kD;
  float* f2e      = ws + off; off += BE * kD;
  float* f2escat  = ws + off; off += BE * kD;
  float* nq       = ws + off; off += BE * kD;
  float* qhid     = ws + off; off += (long)kB * kQ * kD;
  float* qn       = ws + off; off += kB * kD;      // query_node_emb
  float* q2ev     = ws + off; off += kB * kD;
  float* Wf       = ws + off; off += BF;
  float* wtld     = ws + off; off += BF;
  float* wmax     = ws + off; off += kB;
  float* e2fsm    = ws + off; off += BE;
  float* pg       = ws + off; off += BE;
  float* pr       = ws + off; off += BE;
  float* prdiv    = ws + off; off += BE;
  // zero-padded (112 x K) weights
  float* rel_lin_p = ws + off; off += (long)kNP * kD;
  float* head_p[kL]; float* self_p[kL]; float* tail_p[kL];
  float* e2q_p[kL];  float* e2e_p[kL];
  for (int i = 0; i < kL; ++i) {
    head_p[i] = ws + off; off += (long)kNP * kD;
    self_p[i] = ws + off; off += (long)kNP * kD;
    tail_p[i] = ws + off; off += (long)kNP * kD;
    e2q_p[i]  = ws + off; off += (long)kNP * 3 * kD;
    e2e_p[i]  = ws + off; off += (long)kNP * 3 * kD;
  }

  // --- pad all GEMM weights once (tiny) ---
  pad_weight_kernel<<<blks((long)kNP * kD, 256), 256, 0, stream>>>(rel_lin_w, rel_lin_p, kD, kD);
  for (int i = 0; i < kL; ++i) {
    pad_weight_kernel<<<blks((long)kNP * kD, 256), 256, 0, stream>>>(
        head_w + (long)i * kD * kD, head_p[i], kD, kD);
    pad_weight_kernel<<<blks((long)kNP * kD, 256), 256, 0, stream>>>(
        self_w + (long)i * kD * kD, self_p[i], kD, kD);
    pad_weight_kernel<<<blks((long)kNP * kD, 256), 256, 0, stream>>>(
        tail_w + (long)i * kD * kD, tail_p[i], kD, kD);
    pad_weight_kernel<<<blks((long)kNP * 3 * kD, 256), 256, 0, stream>>>(
        e2q_w + (long)i * kD * 3 * kD, e2q_p[i], kD, 3 * kD);
    pad_weight_kernel<<<blks((long)kNP * 3 * kD, 256), 256, 0, stream>>>(
        e2e_w + (long)i * kD * 3 * kD, e2e_p[i], kD, 3 * kD);
  }

  // --- LSTM; q_last initializes query_node_emb ---
  lstm_kernel<<<kB, 128, 0, stream>>>(query_text, word_table, lstm_wih, lstm_whh,
                                      lstm_bih, lstm_bhh, qhid, qn);

  // --- fact_emb = rel_table[kb_fact_rel] @ rel_lin_w.T + b (WMMA, fused gather) ---
  wmma_gemm<EP_BIAS, true, 7><<<gemm_blocks((int)BF), 256, 0, stream>>>(
      rel_table, kb_fact_rel, rel_lin_p, rel_lin_b, fact_emb, (int)BF, kD, kD,
      nullptr, nullptr, nullptr, nullptr, nullptr, 0.f, kF, kE);

  // --- attention -> Wf; W_tilde; e2f_softmax scatter ---
  attn_kernel<<<blks(BF * 32, 256), 256, 0, stream>>>(qhid, fact_emb, query_text, Wf);
  rowmax_kernel<<<kB, 256, 0, stream>>>(Wf, wmax);
  wtilde_kernel<<<blks(BF, 256), 256, 0, stream>>>(Wf, wmax, wtld, (int)BF);
  hipMemsetAsync(e2fsm, 0, BE * sizeof(float), stream);
  scatter_scalar_kernel<<<blks(BF, 256), 256, 0, stream>>>(wtld, kb_head, e2fsm, (int)BF);
  clampmin_kernel<<<blks(BE, 256), 256, 0, stream>>>(e2fsm, kVerySmall, (int)BE);

  // --- entity gather, pagerank init ---
  gather_entity_kernel<<<blks(BE * kD, 256), 256, 0, stream>>>(local_entity, entity_table,
                                                               local, BE * kD);
  copy_kernel<<<blks(BE, 256), 256, 0, stream>>>(q2e_adj, pg, BE);

  // --- L layers ---
  for (int i = 0; i < kL; ++i) {
    const float* q2e_wi  = q2e_w  + (long)i * kD * kD;
    const float* q2e_bi  = q2e_b  + (long)i * kD;
    const float* e2q_bi  = e2q_b  + (long)i * kD;
    const float* e2e_bi  = e2e_b  + (long)i * kD;
    const float* head_bi = head_b + (long)i * kD;
    const float* tail_bi = tail_b + (long)i * kD;
    const float* self_bi = self_b + (long)i * kD;

    // q2e_emb vector (B,D) and pagerank/e2f_softmax ratio
    vecmat_kernel<<<kB, 128, 0, stream>>>(qn, q2e_wi, q2e_bi, q2ev);
    div_kernel<<<blks(BE, 256), 256, 0, stream>>>(pg, e2fsm, prdiv, (int)BE);

    // head projection of entities (WMMA)
    wmma_gemm<EP_BIAS, false, 7><<<gemm_blocks((int)BE), 256, 0, stream>>>(
        local, nullptr, head_p[i], head_bi, headp, (int)BE, kD, kD,
        nullptr, nullptr, nullptr, nullptr, nullptr, 0.f, kF, kE);

    // e2f = relu(fact@self^T + b + gather(headp)) * (wtilde * prdiv[head])  (fused epilogue)
    wmma_gemm<EP_E2F, false, 7><<<gemm_blocks((int)BF), 256, 0, stream>>>(
        fact_emb, nullptr, self_p[i], self_bi, e2f, (int)BF, kD, kD,
        kb_head, headp, wtld, prdiv, nullptr, 0.f, kF, kE);

    // tail projection of facts (WMMA), then scatter-add into (B,E,D)
    wmma_gemm<EP_BIAS, false, 7><<<gemm_blocks((int)BF), 256, 0, stream>>>(
        e2f, nullptr, tail_p[i], tail_bi, tailp, (int)BF, kD, kD,
        nullptr, nullptr, nullptr, nullptr, nullptr, 0.f, kF, kE);
    hipMemsetAsync(f2escat, 0, BE * kD * sizeof(float), stream);
    scatter_rows_kernel<<<blks(BF * kD, 256), 256, 0, stream>>>(tailp, kb_tail, f2escat,
                                                                BF * kD);

    // f2e = relu(local@self^T + b + FACT_SCALE * f2escat)  (fused epilogue)
    wmma_gemm<EP_F2E, false, 7><<<gemm_blocks((int)BE), 256, 0, stream>>>(
        local, nullptr, self_p[i], self_bi, f2e, (int)BE, kD, kD,
        nullptr, nullptr, nullptr, nullptr, f2escat, kFactScale, kF, kE);

    // pagerank update (uses pre-update pg, matching reference order)
    hipMemsetAsync(pr, 0, BE * sizeof(float), stream);
    scatter_pr_kernel<<<blks(BF, 256), 256, 0, stream>>>(pg, kb_head, kb_tail, pr, (int)BF);
    pg_update_kernel<<<blks(BE, 256), 256, 0, stream>>>(pr, pg, (int)BE);

    // nxt = concat(local, broadcast q2ev, f2e)  -> (B,E,3D)
    concat_kernel<<<blks(BE * 3 * kD, 256), 256, 0, stream>>>(local, q2ev, f2e, nxt,
                                                              BE * 3 * kD);

    // query_node_emb = sum_e pg[b,e] * (nxt @ e2q^T + b)[b,e,:]
    wmma_gemm<EP_BIAS, false, 7><<<gemm_blocks((int)BE), 256, 0, stream>>>(
        nxt, nullptr, e2q_p[i], e2q_bi, nq, (int)BE, kD, 3 * kD,
        nullptr, nullptr, nullptr, nullptr, nullptr, 0.f, kF, kE);
    qn_reduce_kernel<<<kB * kD, 256, 0, stream>>>(pg, nq, qn);

    // local = relu(nxt @ e2e^T + b)
    wmma_gemm<EP_RELU, false, 7><<<gemm_blocks((int)BE), 256, 0, stream>>>(
        nxt, nullptr, e2e_p[i], e2e_bi, local, (int)BE, kD, 3 * kD,
        nullptr, nullptr, nullptr, nullptr, nullptr, 0.f, kF, kE);
  }

  // --- final score + entity mask ---
  score_kernel<<<blks(BE, 256), 256, 0, stream>>>(local, score_w, score_b, local_entity,
                                                  out, (int)BE);
}